// Sep_Trans_22789096473424
// MI455X (gfx1250) — compile-verified
//
#include <hip/hip_runtime.h>
#include <stdint.h>

typedef __attribute__((ext_vector_type(2))) float v2f;
typedef __attribute__((ext_vector_type(4))) float v4f;
typedef __attribute__((ext_vector_type(8))) float v8f;

#define NROWS  256   // NV*VL*VL coefficient rows
#define KCELLS 49    // 7x7 table cells
#define KPAD   52    // padded to 13 * 4 (WMMA K=4 steps)
#define KSTEPS 13

// ---------------------------------------------------------------------------
// Kernel 1: transpose lgn [B,32,64,64] -> [B,64,64,32] for contiguous gathers
// ---------------------------------------------------------------------------
__global__ __launch_bounds__(256) void lgn_transpose(const float* __restrict__ in,
                                                     float* __restrict__ out)
{
    __shared__ float t[64 * 33];
    const int blk = blockIdx.x;          // B*H = 512 blocks
    const int b = blk >> 6, y = blk & 63;
    const int tid = threadIdx.x;
#pragma unroll
    for (int r = 0; r < 8; ++r) {
        const int e = tid + r * 256;     // 0..2047
        const int c = e >> 6, x = e & 63;
        t[x * 33 + c] = in[(((size_t)b * 32 + c) * 64 + y) * 64 + x];
    }
    __syncthreads();
#pragma unroll
    for (int r = 0; r < 8; ++r) {
        const int e = tid + r * 256;
        const int x = e >> 5, c = e & 31;
        out[(((size_t)b * 64 + y) * 64 + x) * 32 + c] = t[x * 33 + c];
    }
}

// ---------------------------------------------------------------------------
// Kernel 2: fused gather + bilinear-coef GEMM (WMMA) + per-pixel matvec
//   wave32: 16 pixels / wave, 2 lanes per pixel (lane halves = D row-halves)
// ---------------------------------------------------------------------------
template <bool TR>
__global__ __launch_bounds__(256) void sep_trans_main(
    const float* __restrict__ grid1, const float* __restrict__ grid2,
    const float* __restrict__ lgn,   const float* __restrict__ coefw,
    float* __restrict__ out)
{
    __shared__ __align__(16) float Tlds[NROWS * KPAD];

    // Stage coef table into LDS, re-pitched 49 -> 52 with zeroed pad columns,
    // using CDNA5 async global->LDS loads (ASYNCcnt path).
    {
        const int tid = threadIdx.x;
        for (int idx = tid; idx < NROWS * KPAD; idx += 256) {
            const int ch = idx / KPAD;
            const int c  = idx - ch * KPAD;
            if (c < KCELLS) {
                unsigned ldsoff = (unsigned)(uintptr_t)&Tlds[idx];
                unsigned goff   = (unsigned)((ch * KCELLS + c) * 4);
                asm volatile("global_load_async_to_lds_b32 %0, %1, %2"
                             :: "v"(ldsoff), "v"(goff), "s"(coefw) : "memory");
            } else {
                Tlds[idx] = 0.0f;     // zero K padding (kills spurious row-7 matches)
            }
        }
        asm volatile("s_wait_asynccnt 0" ::: "memory");
    }
    __syncthreads();

    const int tid  = threadIdx.x;
    const int wave = tid >> 5;
    const int lane = tid & 31;
    const int half = lane >> 4;   // D/A/B row-half
    const int pcol = lane & 15;   // pixel column within wave

    const int pix = blockIdx.x * 128 + wave * 16 + pcol;   // 0..32767
    const int b = pix >> 12;
    const int y = (pix >> 6) & 63;
    const int x = pix & 63;

    // nearest-neighbor source location from grid2 (round-half-even like jnp.round)
    const v2f g2 = *(const v2f*)(grid2 + (size_t)pix * 2);
    const float g2x = g2[0], g2y = g2[1];
    const float fxs = rintf((g2x + 1.0f) * 32.0f - 0.5f);
    const float fys = rintf((g2y + 1.0f) * 32.0f - 0.5f);
    const bool valid = (fxs >= 0.0f) && (fxs <= 63.0f) && (fys >= 0.0f) && (fys <= 63.0f);
    const int xsi = (int)fminf(fmaxf(fxs, 0.0f), 63.0f);
    const int ysi = (int)fminf(fmaxf(fys, 0.0f), 63.0f);

    float acc[16];
#pragma unroll
    for (int t = 0; t < 16; ++t) acc[t] = 0.0f;

    for (int k = 0; k < 9; ++k) {               // unfold taps
        const int ki = k / 3;
        const int kj = k - ki * 3;
        const int yy = ysi + ki - 1;
        const int xx = xsi + kj - 1;
        // OOB tap or invalid nearest sample => unfold/gather zero-fill => zero weights
        const bool tap_ok = valid && ((unsigned)yy < 64u) && ((unsigned)xx < 64u);
        const int yyc = yy < 0 ? 0 : (yy > 63 ? 63 : yy);
        const int xxc = xx < 0 ? 0 : (xx > 63 ? 63 : xx);

        const v2f g1 = *(const v2f*)(grid1 + ((((size_t)b * 64 + yyc) * 64) + xxc) * 2);
        // displacement -> 7x7 table coords (grid_sample bilinear, align_corners=False)
        const float ix = (g2x - g1[0] + 1.0f) * 3.5f - 0.5f;
        const float iy = (g2y - g1[1] + 1.0f) * 3.5f - 0.5f;
        const float x0f = floorf(ix), y0f = floorf(iy);
        const float msk = tap_ok ? 1.0f : 0.0f;
        const float fx = ix - x0f, fy = iy - y0f;
        const float wx1 = fx * msk, wx0 = (1.0f - fx) * msk;
        const float wy1 = fy,       wy0 = 1.0f - fy;
        const int cx0 = (int)x0f, cy0 = (int)y0f;

        // Per-lane B-operand values: W[c] for c = 4s + 2*half + q (sparse bilinear
        // weight over 49 cells; out-of-table corners never match -> zero padding).
        float Wc[2 * KSTEPS];
#pragma unroll
        for (int s = 0; s < KSTEPS; ++s) {
#pragma unroll
            for (int q = 0; q < 2; ++q) {
                const int c  = 4 * s + 2 * half + q;
                const int cy = c / 7;
                const int cx = c - cy * 7;
                const float wv = (cy == cy0) ? wy0 : ((cy == cy0 + 1) ? wy1 : 0.0f);
                const float wu = (cx == cx0) ? wx0 : ((cx == cx0 + 1) ? wx1 : 0.0f);
                Wc[2 * s + q] = wv * wu;
            }
        }

#pragma unroll
        for (int mm = 0; mm < 4; ++mm) {        // the 4 vectors (NV)
            float x8[8];
            if (TR) {
                const float* p = lgn + ((((size_t)b * 64 + yyc) * 64) + xxc) * 32 + mm * 8;
                const v4f a0 = *(const v4f*)(p);
                const v4f a1 = *(const v4f*)(p + 4);
#pragma unroll
                for (int j = 0; j < 4; ++j) { x8[j] = a0[j]; x8[4 + j] = a1[j]; }
            } else {
                const float* p = lgn + (((size_t)b * 32 + mm * 8) * 64 + yyc) * 64 + xxc;
#pragma unroll
                for (int j = 0; j < 8; ++j) x8[j] = p[(size_t)j * 4096];
            }
#pragma unroll
            for (int t2 = 0; t2 < 4; ++t2) {    // 4 row-tiles per vector
                const int t = mm * 4 + t2;      // M-tile: rows [16t, 16t+16)
                const float* arow = &Tlds[(t * 16 + pcol) * KPAD + 2 * half];
                v8f dfrag = {0.f, 0.f, 0.f, 0.f, 0.f, 0.f, 0.f, 0.f};
#pragma unroll
                for (int s = 0; s < KSTEPS; ++s) {
                    const v2f a  = *(const v2f*)(arow + 4 * s);   // A[M=row, K=c..c+1]
                    const v2f bb = { Wc[2 * s], Wc[2 * s + 1] };  // B[K=c..c+1, N=pcol]
                    // D[r, p] += sum_c T[r, c] * W[c, p]   (interpolated coefficients)
                    dfrag = __builtin_amdgcn_wmma_f32_16x16x4_f32(
                        false, a, false, bb, (short)0, dfrag, false, false);
                }
                // Lane (half,pcol) holds Ceff rows 16t+8*half+j (j = vgpr) for its
                // own pixel => contract directly against x8 (channel pair 2t+half).
                float dot = 0.0f;
#pragma unroll
                for (int j = 0; j < 8; ++j) dot += dfrag[j] * x8[j];
                acc[t] += dot;
            }
        }
    }

    // out channel = (m*8 + i) = 2t + half; stores coalesce across pcol lanes
#pragma unroll
    for (int t = 0; t < 16; ++t) {
        const int P = 2 * t + half;
        out[(((size_t)b * 32 + P) << 12) + y * 64 + x] = acc[t];
    }
}

// ---------------------------------------------------------------------------
extern "C" void kernel_launch(void* const* d_in, const int* in_sizes, int n_in,
                              void* d_out, int out_size, void* d_ws, size_t ws_size,
                              hipStream_t stream)
{
    (void)in_sizes; (void)n_in; (void)out_size;
    const float* grid1 = (const float*)d_in[0];
    const float* grid2 = (const float*)d_in[1];
    // d_in[2] = theta: unused by the reference forward
    const float* lgn   = (const float*)d_in[3];
    const float* coefw = (const float*)d_in[4];
    float* out = (float*)d_out;

    const size_t lgnT_bytes = (size_t)8 * 64 * 64 * 32 * sizeof(float);
    if (d_ws && ws_size >= lgnT_bytes) {
        float* lgnT = (float*)d_ws;
        lgn_transpose<<<512, 256, 0, stream>>>(lgn, lgnT);
        sep_trans_main<true><<<256, 256, 0, stream>>>(grid1, grid2, lgnT, coefw, out);
    } else {
        sep_trans_main<false><<<256, 256, 0, stream>>>(grid1, grid2, lgn, coefw, out);
    }
}